// StarLoss_20478404067923
// MI455X (gfx1250) — compile-verified
//
#include <hip/hip_runtime.h>

// ---------------------------------------------------------------------------
// StarLoss on MI455X (gfx1250, wave32).
// Bandwidth-bound 5-way global reduction (537 MB streamed once -> ~23 us at
// 23.3 TB/s). Main loop: b128/b64 non-temporal loads + VALU (v_rsq_f32 for
// all reciprocal square roots -> no IEEE div sequences). Epilogue: wave32
// reduction via V_WMMA_F32_16X16X4_F32 (A = accumulators, B = ones -> row
// sums), closed with one xor-16 shuffle; LDS across waves; 2nd kernel across
// blocks. Deterministic (no float atomics), graph-capture safe.
// ---------------------------------------------------------------------------

typedef float v2f __attribute__((ext_vector_type(2)));
typedef float v4f __attribute__((ext_vector_type(4)));
typedef float v8f __attribute__((ext_vector_type(8)));

#define STAR_OMEGA 1.0f
#define STAR_EPS   1e-5f

// ~1-ulp hardware rsqrt (V_RSQ_F32); avoids the full IEEE divide sequence.
__device__ __forceinline__ float fast_rsqrt(float x) {
#if __has_builtin(__builtin_amdgcn_rsqf)
  return __builtin_amdgcn_rsqf(x);
#else
  return 1.0f / sqrtf(x);
#endif
}

// Sum one f32 value across all 32 lanes of a wave.
// V_WMMA_F32_16X16X4_F32: D = A(16x4) x B(4x16) + C.
// A layout (32-bit, 16x4): lane l<16 -> A[l][0]=VGPR0, A[l][1]=VGPR1;
//                          lane l>=16 -> A[l-16][2]=VGPR0, A[l-16][3]=VGPR1.
// With A-VGPR0 = x, A-VGPR1 = 0, B = ones: D[m][n] = x_m + x_{m+16}.
// D layout: lane l<16 VGPR v = D[v][l]; lane l>=16 VGPR v = D[v+8][l-16].
// summing the 8 D registers -> half-wave sum; xor-16 shuffle adds the other half.
__device__ __forceinline__ float wave_sum32(float x) {
#if __has_builtin(__builtin_amdgcn_wmma_f32_16x16x4_f32)
  v2f a; a[0] = x;    a[1] = 0.0f;
  v2f b; b[0] = 1.0f; b[1] = 1.0f;
  v8f c = {};
  v8f d = __builtin_amdgcn_wmma_f32_16x16x4_f32(
      /*neg_a=*/false, a, /*neg_b=*/false, b,
      /*c_mod=*/(short)0, c, /*reuse_a=*/false, /*reuse_b=*/false);
  float s = ((d[0] + d[1]) + (d[2] + d[3])) + ((d[4] + d[5]) + (d[6] + d[7]));
  s += __shfl_xor(s, 16, 32);
  return s;
#else
  for (int m = 16; m >= 1; m >>= 1) x += __shfl_xor(x, m, 32);
  return x;
#endif
}

// Reduce 5 per-thread accumulators to 5 per-block sums (blockDim.x == 256,
// 8 waves). All 256 threads must be live (EXEC all-ones for WMMA).
__device__ __forceinline__ void block_reduce5(float acc[5], float out[5]) {
  __shared__ float red[8][5];
  const int wave = threadIdx.x >> 5;
  const int lane = threadIdx.x & 31;
  float w0 = wave_sum32(acc[0]);
  float w1 = wave_sum32(acc[1]);
  float w2 = wave_sum32(acc[2]);
  float w3 = wave_sum32(acc[3]);
  float w4 = wave_sum32(acc[4]);
  if (lane == 0) {
    red[wave][0] = w0; red[wave][1] = w1; red[wave][2] = w2;
    red[wave][3] = w3; red[wave][4] = w4;
  }
  __syncthreads();
  if (threadIdx.x == 0) {
    for (int j = 0; j < 5; ++j) {
      float s = 0.0f;
      for (int w = 0; w < 8; ++w) s += red[w][j];
      out[j] = s;
    }
  }
}

__global__ __launch_bounds__(256) void star_partial_kernel(
    const float* __restrict__ pred, const float* __restrict__ cov,
    const float* __restrict__ target, float* __restrict__ partial,
    long long n) {
  const v4f* __restrict__ cov4 = (const v4f*)cov;
  const v2f* __restrict__ prd2 = (const v2f*)pred;
  const v2f* __restrict__ tgt2 = (const v2f*)target;

  float acc[5] = {0.0f, 0.0f, 0.0f, 0.0f, 0.0f};  // S1 S2 M1 M2 T

  const long long stride = (long long)gridDim.x * blockDim.x;
#pragma unroll 4
  for (long long i = (long long)blockIdx.x * blockDim.x + threadIdx.x; i < n;
       i += stride) {
    // Streamed exactly once and > L2 capacity -> non-temporal loads.
    v4f cv = __builtin_nontemporal_load(&cov4[i]);  // [a, b, b, d]
    v2f pp = __builtin_nontemporal_load(&prd2[i]);
    v2f tt = __builtin_nontemporal_load(&tgt2[i]);

    const float a  = cv[0];
    const float b  = cv[1];
    const float dd = cv[3];

    const float mean  = 0.5f * (a + dd);
    const float hdiff = 0.5f * (a - dd);
    const float delta = sqrtf(hdiff * hdiff + b * b);
    const float lam1  = mean + delta;
    const float lam2  = mean - delta;

    const bool  diag = fabsf(b) < 1e-12f;
    float vx = diag ? (a >= dd ? 1.0f : 0.0f) : b;
    float vy = diag ? (a >= dd ? 0.0f : 1.0f) : (lam1 - a);
    const float invn = fast_rsqrt(vx * vx + vy * vy);
    vx *= invn;
    vy *= invn;

    const float dx = pp[0] - tt[0];
    const float dy = pp[1] - tt[1];
    const float p1 = vx * dx + vy * dy;
    const float p2 = vx * dy - vy * dx;

    acc[0] = fmaf(p1, p1, acc[0]);
    acc[1] = fmaf(p2, p2, acc[1]);
    acc[2] += fast_rsqrt(lam1) + STAR_EPS;
    acc[3] += fast_rsqrt(lam2) + STAR_EPS;
    acc[4] += a + dd;  // lam1 + lam2 == trace
  }

  float bsum[5];
  block_reduce5(acc, bsum);
  if (threadIdx.x == 0) {
    float* p = partial + 5ll * blockIdx.x;
    p[0] = bsum[0]; p[1] = bsum[1]; p[2] = bsum[2];
    p[3] = bsum[3]; p[4] = bsum[4];
  }
}

__global__ __launch_bounds__(256) void star_final_kernel(
    const float* __restrict__ partial, int nparts, float* __restrict__ out,
    float inv_n) {
  float acc[5] = {0.0f, 0.0f, 0.0f, 0.0f, 0.0f};
  for (int i = threadIdx.x; i < nparts; i += 256) {
    const float* p = partial + 5 * i;
    acc[0] += p[0]; acc[1] += p[1]; acc[2] += p[2];
    acc[3] += p[3]; acc[4] += p[4];
  }
  float s[5];
  block_reduce5(acc, s);
  if (threadIdx.x == 0) {
    // mean(loss) = S1*mean(coef1) + S2*mean(coef2) + 0.5*omega*mean(trace)
    out[0] = (s[0] * s[2] + s[1] * s[3] + 0.5f * STAR_OMEGA * s[4]) * inv_n;
  }
}

extern "C" void kernel_launch(void* const* d_in, const int* in_sizes, int n_in,
                              void* d_out, int out_size, void* d_ws,
                              size_t ws_size, hipStream_t stream) {
  (void)n_in; (void)out_size;
  const float* pred   = (const float*)d_in[0];
  const float* cov    = (const float*)d_in[1];
  const float* target = (const float*)d_in[2];
  float* out = (float*)d_out;
  float* partial = (float*)d_ws;

  const long long n = (long long)in_sizes[0] / 2;  // B*N elements

  int nblocks = 1024;  // 262144 threads -> 64 iters x 32 B each; saturates HBM
  const long long maxb = (long long)(ws_size / (5 * sizeof(float)));
  if (maxb > 0 && nblocks > maxb) nblocks = (int)maxb;
  if (nblocks < 1) nblocks = 1;

  star_partial_kernel<<<nblocks, 256, 0, stream>>>(pred, cov, target, partial,
                                                   n);
  star_final_kernel<<<1, 256, 0, stream>>>(partial, nblocks, out,
                                           (float)(1.0 / (double)n));
}